// LineOptimizer_53824530153873
// MI455X (gfx1250) — compile-verified
//
#include <hip/hip_runtime.h>

#define SQRT3F 1.7320508075688772f
#define WSZ 32
#define MAXLC 1024

typedef float v2f __attribute__((ext_vector_type(2)));
typedef float v8f __attribute__((ext_vector_type(8)));

// ---------------------------------------------------------------------------
// Kernel 1: base = P/(sqrt3*pf); block partial sums of I0 = base/ue and I0*x.
// grid = L*Bb blocks, 256 threads, 4 contiguous elements per thread.
// ---------------------------------------------------------------------------
template<int C>
__global__ __launch_bounds__(256)
void k_init(const float* __restrict__ P, const float* __restrict__ pf,
            const float* __restrict__ x, const float* __restrict__ ue,
            float* __restrict__ baseOut,
            float* __restrict__ sumI, float* __restrict__ sumIx,
            int N, int Bb)
{
    const int R = 4;
    int l = blockIdx.x / Bb;
    int b = blockIdx.x % Bb;
    int t = threadIdx.x;
    int lane = t & (WSZ - 1);
    int wv = t >> 5;
    long long e0 = (long long)l * N + (long long)b * 1024 + t * R;

    float4 P4  = *(const float4*)(P + e0);
    float4 pf4 = *(const float4*)(pf + e0);
    float4 x4  = *(const float4*)(x + e0);
    float bse[R] = {P4.x / (SQRT3F * pf4.x), P4.y / (SQRT3F * pf4.y),
                    P4.z / (SQRT3F * pf4.z), P4.w / (SQRT3F * pf4.w)};
    float xx[R]  = {x4.x, x4.y, x4.z, x4.w};
    *(float4*)(baseOut + e0) = make_float4(bse[0], bse[1], bse[2], bse[3]);

    float uec[C];
#pragma unroll
    for (int c = 0; c < C; c++) uec[c] = ue[c];

    float sI[C], sIx[C];
#pragma unroll
    for (int c = 0; c < C; c++) { sI[c] = 0.f; sIx[c] = 0.f; }
#pragma unroll
    for (int e = 0; e < R; e++)
#pragma unroll
        for (int c = 0; c < C; c++) {
            float Ii = bse[e] / uec[c];
            sI[c] += Ii; sIx[c] += Ii * xx[e];
        }

    // butterfly wave reduce + cross-wave LDS reduce
#pragma unroll
    for (int c = 0; c < C; c++) {
        float v = sI[c], v2 = sIx[c];
        for (int d = 16; d >= 1; d >>= 1) { v += __shfl_xor(v, d, WSZ); v2 += __shfl_xor(v2, d, WSZ); }
        sI[c] = v; sIx[c] = v2;
    }
    __shared__ float rsI[8][C], rsIx[8][C];
    if (lane == 0)
#pragma unroll
        for (int c = 0; c < C; c++) { rsI[wv][c] = sI[c]; rsIx[wv][c] = sIx[c]; }
    __syncthreads();
    if (t == 0) {
        long long cbase = ((long long)l * Bb + b) * C;
#pragma unroll
        for (int c = 0; c < C; c++) {
            float a = 0.f, a2 = 0.f;
            for (int w = 0; w < 8; w++) { a += rsI[w][c]; a2 += rsIx[w][c]; }
            sumI[cbase + c] = a; sumIx[cbase + c] = a2;
        }
    }
}

// ---------------------------------------------------------------------------
// Kernel 2 ("mid", 1 workgroup): scan block partials -> exclusive carries,
// line totals, cross-line serial chain (childI, v_line), last sweep -> output.
// Uses V_WMMA_F32_16X16X4_F32 (exact f32) for group-of-4 sums when Bb==64:
//   A[m,k] = w[4m+k] (documented layout: lanes 0-15 hold K=0,1; 16-31 K=2,3),
//   B = ones  =>  D[m,n] = group sum m (constant over n).
// ---------------------------------------------------------------------------
template<int C>
__global__ __launch_bounds__(256)
void k_mid(const float* __restrict__ sumI, const float* __restrict__ sumIx,
           const float* __restrict__ x, const float* __restrict__ rho,
           const float* __restrict__ ue,
           float* __restrict__ carries, float* __restrict__ lsCur,
           float* __restrict__ outp,
           int L, int N, int Bb, int isLast)
{
    __shared__ float ItotL[MAXLC], IxtL[MAXLC], dUeL[MAXLC], SL[MAXLC];
    int t = threadIdx.x, lane = t & (WSZ - 1), wv = t >> 5;
    int nw = blockDim.x >> 5;
    int Pn = L * C * 2;

#if __has_builtin(__builtin_amdgcn_wmma_f32_16x16x4_f32)
    if (Bb == 64) {
        for (int p = wv; p < Pn; p += nw) {   // uniform per wave -> EXEC all-ones at WMMA
            int which = p & 1; int lc = p >> 1; int l = lc / C; int c = lc % C;
            const float* src = which ? sumIx : sumI;
            long long rowb = ((long long)l * Bb) * C + c;   // w[b] at rowb + b*C
            int m = lane & 15, h = lane >> 4;
            int b0 = 4 * m + 2 * h;
            v2f A; A.x = src[rowb + (long long)b0 * C]; A.y = src[rowb + (long long)(b0 + 1) * C];
            v2f Bv; Bv.x = 1.f; Bv.y = 1.f;
            v8f Cz = {0.f, 0.f, 0.f, 0.f, 0.f, 0.f, 0.f, 0.f};
            v8f D = __builtin_amdgcn_wmma_f32_16x16x4_f32(false, A, false, Bv,
                                                          (short)0, Cz, false, false);
            // D layout: d[i] = groupsum(i + 8*h); gather other half via xor-16 shuffle
            float gsAll[16];
#pragma unroll
            for (int i = 0; i < 8; i++) {
                float own = D[i];
                float oth = __shfl_xor(own, 16, WSZ);
                gsAll[i]     = (h == 0) ? own : oth;
                gsAll[i + 8] = (h == 0) ? oth : own;
            }
            int j = lane, g = j >> 1;
            float pre = 0.f, tot = 0.f;
#pragma unroll
            for (int i = 0; i < 16; i++) { if (i < g) pre += gsAll[i]; tot += gsAll[i]; }
            int i1 = (2 * j - 2 < 0) ? 0 : 2 * j - 2;
            int i2 = (2 * j - 1 < 0) ? 0 : 2 * j - 1;
            float w1 = src[rowb + (long long)i1 * C];
            float w2 = src[rowb + (long long)i2 * C];
            float w3 = src[rowb + (long long)(2 * j) * C];
            float ce = pre + ((j & 1) ? (w1 + w2) : 0.f);   // exclusive carry, block 2j
            float co = ce + w3;                              // exclusive carry, block 2j+1
            long long cb = ((long long)l * Bb + 2 * j) * C + c;
            carries[cb * 2 + which] = ce;
            carries[(cb + C) * 2 + which] = co;
            if (lane == 31) { if (which) IxtL[lc] = tot; else ItotL[lc] = tot; }
        }
    } else
#endif
    {
        for (int p = t; p < Pn; p += blockDim.x) {
            int which = p & 1; int lc = p >> 1; int l = lc / C; int c = lc % C;
            const float* src = which ? sumIx : sumI;
            float run = 0.f;
            for (int b = 0; b < Bb; b++) {
                long long cb = ((long long)l * Bb + b) * C + c;
                carries[cb * 2 + which] = run;
                run += src[cb];
            }
            if (which) IxtL[lc] = run; else ItotL[lc] = run;
        }
    }
    __syncthreads();

    int LC = L * C;
    for (int q = t; q < LC; q += blockDim.x) {
        int l = q / C;
        float childI = (l < L - 1) ? ItotL[q + C] : 0.f;
        float dUe = rho[q] * (IxtL[q] + x[(long long)l * N + (N - 1)] * childI);
        dUeL[q] = dUe;
        SL[q] = ItotL[q] + childI;
    }
    __syncthreads();
    for (int q = t; q < LC; q += blockDim.x) {
        int l = q / C, c = q % C;
        float pref = 0.f;
        for (int j = 0; j < l; j++) pref += dUeL[j * C + c];
        float vline = ue[c] - pref;
        lsCur[q * 2 + 0] = vline;
        lsCur[q * 2 + 1] = SL[q];
        if (isLast) outp[q] = (1.f - (vline - dUeL[q]) / ue[c]) * 100.f;
    }
}

// ---------------------------------------------------------------------------
// Kernel 3 ("sweep"): reconstruct I_k = base/(vline_{k-1} - dU16_{k-1}),
// in-block prefix (wave32 Kogge-Stone scan + cross-wave LDS), dUx_k -> fp16,
// v_load_k, I_{k+1}, block partial sums of I_{k+1} for the next mid kernel.
// ---------------------------------------------------------------------------
template<int C>
__global__ __launch_bounds__(256)
void k_sweep(const float* __restrict__ base, const float* __restrict__ x,
             const float* __restrict__ rho, const float* __restrict__ ue,
             _Float16* __restrict__ du,
             const float* __restrict__ carries,
             const float* __restrict__ lsPrev, const float* __restrict__ lsCur,
             float* __restrict__ sumI, float* __restrict__ sumIx,
             int N, int Bb, int isFirst, int storeDu)
{
    const int R = 4;
    int l = blockIdx.x / Bb;
    int b = blockIdx.x % Bb;
    int t = threadIdx.x;
    int lane = t & (WSZ - 1);
    int wv = t >> 5;
    long long e0 = (long long)l * N + (long long)b * 1024 + t * R;

    float4 b4 = *(const float4*)(base + e0);
    float4 x4 = *(const float4*)(x + e0);
    float bse[R] = {b4.x, b4.y, b4.z, b4.w};
    float xx[R]  = {x4.x, x4.y, x4.z, x4.w};

    float vlp[C], vlk[C], Sk[C], rh[C];
#pragma unroll
    for (int c = 0; c < C; c++) {
        long long q = (long long)l * C + c;
        vlk[c] = lsCur[q * 2 + 0];
        Sk[c]  = lsCur[q * 2 + 1];
        rh[c]  = rho[q];
        vlp[c] = isFirst ? ue[c] : lsPrev[q * 2 + 0];
    }

    _Float16 duLoc[R * C];
    if (!isFirst) {
        const _Float16* p = du + e0 * C;
#pragma unroll
        for (int i = 0; i < R * C; i++) duLoc[i] = p[i];
    }

    float I[R][C], sI[C], sIx[C];
#pragma unroll
    for (int c = 0; c < C; c++) { sI[c] = 0.f; sIx[c] = 0.f; }
#pragma unroll
    for (int e = 0; e < R; e++)
#pragma unroll
        for (int c = 0; c < C; c++) {
            float vp = isFirst ? vlp[c] : (vlp[c] - (float)duLoc[e * C + c]);
            float Ii = bse[e] / vp;
            I[e][c] = Ii; sI[c] += Ii; sIx[c] += Ii * xx[e];
        }

    // wave32 inclusive scan of per-thread totals
    float inI[C], inIx[C];
#pragma unroll
    for (int c = 0; c < C; c++) {
        float v = sI[c], v2 = sIx[c];
        for (int d = 1; d < WSZ; d <<= 1) {
            float y = __shfl_up(v, d, WSZ);
            float y2 = __shfl_up(v2, d, WSZ);
            if (lane >= d) { v += y; v2 += y2; }
        }
        inI[c] = v; inIx[c] = v2;
    }
    __shared__ float wsI[8][C], wsIx[8][C];
    if (lane == WSZ - 1)
#pragma unroll
        for (int c = 0; c < C; c++) { wsI[wv][c] = inI[c]; wsIx[wv][c] = inIx[c]; }
    __syncthreads();

    long long cbase = ((long long)l * Bb + b) * C;
    float runI[C], runIx[C], nsI[C], nsIx[C];
#pragma unroll
    for (int c = 0; c < C; c++) {
        float o = 0.f, o2 = 0.f;
        for (int w = 0; w < 8; w++)
            if (w < wv) { o += wsI[w][c]; o2 += wsIx[w][c]; }
        runI[c]  = carries[(cbase + c) * 2 + 0] + o  + (inI[c]  - sI[c]);
        runIx[c] = carries[(cbase + c) * 2 + 1] + o2 + (inIx[c] - sIx[c]);
        nsI[c] = 0.f; nsIx[c] = 0.f;
    }

    _Float16 duNew[R * C];
#pragma unroll
    for (int e = 0; e < R; e++)
#pragma unroll
        for (int c = 0; c < C; c++) {
            runI[c]  += I[e][c];              // inclusive cs_I at this element
            runIx[c] += I[e][c] * xx[e];      // inclusive cs_Ix
            float dU = rh[c] * (runIx[c] + xx[e] * (Sk[c] - runI[c]));
            _Float16 hh = (_Float16)dU;
            duNew[e * C + c] = hh;
            float vload = vlk[c] - (float)hh;
            float In = bse[e] / vload;
            nsI[c] += In; nsIx[c] += In * xx[e];
        }
    if (storeDu) {
        _Float16* p = du + e0 * C;
#pragma unroll
        for (int i = 0; i < R * C; i++) p[i] = duNew[i];
    }

    // block reduce next-sweep partial sums
#pragma unroll
    for (int c = 0; c < C; c++) {
        float v = nsI[c], v2 = nsIx[c];
        for (int d = 16; d >= 1; d >>= 1) { v += __shfl_xor(v, d, WSZ); v2 += __shfl_xor(v2, d, WSZ); }
        nsI[c] = v; nsIx[c] = v2;
    }
    __shared__ float rsI[8][C], rsIx[8][C];
    if (lane == 0)
#pragma unroll
        for (int c = 0; c < C; c++) { rsI[wv][c] = nsI[c]; rsIx[wv][c] = nsIx[c]; }
    __syncthreads();
    if (t == 0) {
#pragma unroll
        for (int c = 0; c < C; c++) {
            float a = 0.f, a2 = 0.f;
            for (int w = 0; w < 8; w++) { a += rsI[w][c]; a2 += rsIx[w][c]; }
            sumI[cbase + c] = a; sumIx[cbase + c] = a2;
        }
    }
}

// ---------------------------------------------------------------------------
// Host-side pipeline
// ---------------------------------------------------------------------------
template<int C>
static void runAll(const float* rho, const float* P, const float* pf,
                   const float* x, const float* ue, float* out,
                   void* ws, size_t ws_size, int L, int N, hipStream_t stream)
{
    int Bb = N / 1024;
    long long LN = (long long)L * N;
    size_t off = 0;
    auto carve = [&](size_t bytes) -> char* {
        char* p = (char*)ws + off;
        off += (bytes + 255) & ~(size_t)255;
        return p;
    };
    float*    base     = (float*)carve((size_t)LN * 4);
    _Float16* du       = (_Float16*)carve((size_t)LN * C * 2);
    float*    sumI     = (float*)carve((size_t)L * Bb * C * 4);
    float*    sumIx    = (float*)carve((size_t)L * Bb * C * 4);
    float*    carries  = (float*)carve((size_t)L * Bb * C * 2 * 4);
    float*    linescal = (float*)carve((size_t)5 * L * C * 2 * 4);
    if (off > ws_size) return;
    if (L * C > MAXLC) return;

    dim3 grid(L * Bb), blk(256);
    k_init<C><<<grid, blk, 0, stream>>>(P, pf, x, ue, base, sumI, sumIx, N, Bb);
    for (int k = 0; k < 5; k++) {
        float* lsCur = linescal + (size_t)k * L * C * 2;
        k_mid<C><<<1, 256, 0, stream>>>(sumI, sumIx, x, rho, ue, carries, lsCur, out,
                                        L, N, Bb, (k == 4) ? 1 : 0);
        if (k < 4) {
            const float* lsPrev = k ? (linescal + (size_t)(k - 1) * L * C * 2) : nullptr;
            k_sweep<C><<<grid, blk, 0, stream>>>(base, x, rho, ue, du, carries,
                                                 lsPrev, lsCur, sumI, sumIx,
                                                 N, Bb, (k == 0) ? 1 : 0, (k < 3) ? 1 : 0);
        }
    }
}

extern "C" void kernel_launch(void* const* d_in, const int* in_sizes, int n_in,
                              void* d_out, int out_size, void* d_ws, size_t ws_size,
                              hipStream_t stream)
{
    (void)n_in; (void)out_size;
    const float* rho = (const float*)d_in[0];
    const float* P   = (const float*)d_in[1];
    const float* pf  = (const float*)d_in[2];
    const float* x   = (const float*)d_in[3];
    const float* ue  = (const float*)d_in[4];
    float* out = (float*)d_out;

    int C = in_sizes[4];
    if (C <= 0) return;
    int L = in_sizes[0] / C;
    if (L <= 0) return;
    int N = in_sizes[1] / L;
    if (N % 1024) return;   // unsupported shape for this pipeline

    if (C == 4)      runAll<4>(rho, P, pf, x, ue, out, d_ws, ws_size, L, N, stream);
    else if (C == 1) runAll<1>(rho, P, pf, x, ue, out, d_ws, ws_size, L, N, stream);
    else if (C == 2) runAll<2>(rho, P, pf, x, ue, out, d_ws, ws_size, L, N, stream);
    else if (C == 8) runAll<8>(rho, P, pf, x, ue, out, d_ws, ws_size, L, N, stream);
}